// AFF_MambaLayer_19241453486091
// MI455X (gfx1250) — compile-verified
//
#include <hip/hip_runtime.h>
#include <hip/hip_bf16.h>

// Problem constants (from the reference)
#define LTOT 8192   // D*H*W
#define CCH  96     // C
#define DINC 192    // EXPAND*C
#define NST  16     // D_STATE
#define RNK  6      // DT_RANK
#define PLD  48     // padded width for xproj output (38 valid -> 3 wmma tiles)
#define TCH  64     // scan chunk length (8192 -> 128 chunks, 1024 -> 16 chunks)

typedef float v2f __attribute__((ext_vector_type(2)));
typedef float v8f __attribute__((ext_vector_type(8)));

__device__ __forceinline__ float silu_f(float v) { return v / (1.f + __expf(-v)); }
__device__ __forceinline__ float sigm_f(float v) { return 1.f / (1.f + __expf(-v)); }

// ---------------------------------------------------------------------------
// LayerNorm over channel dim; x is (C, L), output xn is (L, C)
// ---------------------------------------------------------------------------
__global__ void ln_kernel(const float* __restrict__ x, const float* __restrict__ gam,
                          const float* __restrict__ bet, float* __restrict__ xn) {
  int l = blockIdx.x * blockDim.x + threadIdx.x;
  if (l >= LTOT) return;
  float mu = 0.f;
  for (int c = 0; c < CCH; ++c) mu += x[(size_t)c * LTOT + l];
  mu *= (1.f / CCH);
  float var = 0.f;
  for (int c = 0; c < CCH; ++c) { float d = x[(size_t)c * LTOT + l] - mu; var += d * d; }
  var *= (1.f / CCH);
  float rs = rsqrtf(var + 1e-5f);
  for (int c = 0; c < CCH; ++c)
    xn[(size_t)l * CCH + c] = (x[(size_t)c * LTOT + l] - mu) * rs * gam[c] + bet[c];
}

// ---------------------------------------------------------------------------
// Generic fp32 WMMA GEMM: Y[M x N] = X[M x K] * W[Nvalid x K]^T
// One wave per 16x16 tile; K stepped by 4 via V_WMMA_F32_16X16X4_F32.
// Branchless N-padding: clamp row, scale fragment by 0/1 mask (only taken
// when Nvalid < N, which is wave-uniform -> no EXEC churn in the hot loop).
// ---------------------------------------------------------------------------
__global__ void gemm_wmma_f32(const float* __restrict__ X, const float* __restrict__ W,
                              float* __restrict__ Y, int M, int N, int K, int Nvalid) {
  int wavesPerBlock = blockDim.x >> 5;
  int waveId = blockIdx.x * wavesPerBlock + (threadIdx.x >> 5);
  int lane = threadIdx.x & 31;
  int ntiles = N >> 4;
  int mtile = waveId / ntiles;
  int ntile = waveId % ntiles;
  if (mtile * 16 >= M) return;              // wave-uniform exit: EXEC full at WMMA
  int mb = mtile * 16, nb = ntile * 16;
  int hi = lane >> 4;
  int lo = lane & 15;
  v8f c = {0.f, 0.f, 0.f, 0.f, 0.f, 0.f, 0.f, 0.f};
  const float* xrow = X + (size_t)(mb + lo) * K + 2 * hi;
  int wr = (nb + lo) < Nvalid ? (nb + lo) : 0;  // clamped row (always in-bounds)
  float wm = (nb + lo) < Nvalid ? 1.f : 0.f;
  const float* wrow = W + (size_t)wr * K + 2 * hi;
  if (Nvalid == N) {                        // common case: no masking at all
    for (int kb = 0; kb < K; kb += 4) {
      float2 av = *(const float2*)(xrow + kb);
      float2 bv = *(const float2*)(wrow + kb);
      v2f a; a.x = av.x; a.y = av.y;
      v2f b; b.x = bv.x; b.y = bv.y;
      c = __builtin_amdgcn_wmma_f32_16x16x4_f32(false, a, false, b, (short)0, c, false, false);
    }
  } else {
    for (int kb = 0; kb < K; kb += 4) {
      float2 av = *(const float2*)(xrow + kb);
      float2 bv = *(const float2*)(wrow + kb);
      v2f a; a.x = av.x; a.y = av.y;
      v2f b; b.x = bv.x * wm; b.y = bv.y * wm;
      c = __builtin_amdgcn_wmma_f32_16x16x4_f32(false, a, false, b, (short)0, c, false, false);
    }
  }
  float* yb = Y + (size_t)(mb + 8 * hi) * N + nb + lo;
#pragma unroll
  for (int i = 0; i < 8; ++i) yb[(size_t)i * N] = c[i];
}

// ---------------------------------------------------------------------------
// Causal depthwise conv (k=4) + SiLU on u = xz[:, :192]; also g = silu(z)
// ---------------------------------------------------------------------------
__global__ void conv_silu_kernel(const float* __restrict__ xz, const float* __restrict__ cw,
                                 const float* __restrict__ cb, float* __restrict__ uc,
                                 float* __restrict__ g) {
  long idx = (long)blockIdx.x * blockDim.x + threadIdx.x;
  if (idx >= (long)LTOT * DINC) return;
  long l = idx / DINC; int d = (int)(idx % DINC);
  float acc = cb[d];
#pragma unroll
  for (int k = 0; k < 4; ++k) {
    long j = l - 3 + k;
    if (j >= 0) acc += cw[d * 4 + k] * xz[j * (2 * DINC) + d];
  }
  uc[idx] = silu_f(acc);
  float zv = xz[l * (2 * DINC) + DINC + d];
  g[idx] = silu_f(zv);
}

// ---------------------------------------------------------------------------
// dt = softplus(p[:, :6] @ dt_w^T + dt_b); p: (L, 48), dt: (L, 192)
// ---------------------------------------------------------------------------
__global__ void dt_kernel(const float* __restrict__ p, const float* __restrict__ dtw,
                          const float* __restrict__ dtb, float* __restrict__ dt) {
  long idx = (long)blockIdx.x * blockDim.x + threadIdx.x;
  if (idx >= (long)LTOT * DINC) return;
  long l = idx / DINC; int d = (int)(idx % DINC);
  float acc = dtb[d];
#pragma unroll
  for (int r = 0; r < RNK; ++r) acc += p[l * PLD + r] * dtw[d * RNK + r];
  dt[idx] = (acc > 20.f) ? acc : log1pf(__expf(acc));
}

// ---------------------------------------------------------------------------
// Chunked parallel scan (h[t] = dA[t]*h[t-1] + dBu[t] is associative).
// Pass 1 (summary): per (chunk, d): Aprod[n] = prod(dA), hend[n] = local scan end.
// Pass 2 (prefix):  tiny sequential scan over chunk summaries -> h0 per chunk
//                   (written in-place into chH).
// Pass 3 (final):   re-run each chunk seeded with h0, emit y (+ u*Dp).
// Chunks are indexed in *traversal* order; reverse=1 walks t = T-1..0.
// B/C rows for the chunk are staged in LDS (one barrier per chunk).
// Total work is independent of TCH, so small chunks = more parallelism free.
// ---------------------------------------------------------------------------
__global__ void scan_summary_kernel(const float* __restrict__ uc, const float* __restrict__ dt,
                                    const float* __restrict__ p, const float* __restrict__ A_log,
                                    float* __restrict__ chA, float* __restrict__ chH,
                                    int T, int cps, int reverse) {
  __shared__ float shB[TCH * NST];          // B rows for this chunk: 4 KB
  int gc = blockIdx.x;                      // global chunk id
  int s = gc / cps, c = gc % cps;
  long seqbase = (long)s * T;
  for (int idx = threadIdx.x; idx < TCH * NST; idx += blockDim.x) {
    int i = idx >> 4, col = idx & 15;
    int pos = c * TCH + i;
    int t = reverse ? (T - 1 - pos) : pos;
    shB[idx] = p[(seqbase + t) * PLD + RNK + col];
  }
  __syncthreads();
  int d = threadIdx.x;                      // blockDim.x == DINC
  float A[NST], h[NST], Aac[NST];
#pragma unroll
  for (int n = 0; n < NST; ++n) {
    A[n] = -__expf(A_log[d * NST + n]);
    h[n] = 0.f; Aac[n] = 1.f;
  }
  for (int i = 0; i < TCH; ++i) {
    int pos = c * TCH + i;
    int t = reverse ? (T - 1 - pos) : pos;
    long l = seqbase + t;
    float dtv = dt[l * DINC + d];
    float uv = uc[l * DINC + d];
    float du = dtv * uv;
    const float* bs = shB + i * NST;
#pragma unroll
    for (int n = 0; n < NST; ++n) {
      float dA = __expf(dtv * A[n]);
      h[n] = dA * h[n] + du * bs[n];
      Aac[n] *= dA;
    }
  }
  long o = (long)gc * DINC * NST + (long)d * NST;
#pragma unroll
  for (int n = 0; n < NST; ++n) { chA[o + n] = Aac[n]; chH[o + n] = h[n]; }
}

__global__ void scan_prefix_kernel(const float* __restrict__ chA, float* __restrict__ chH,
                                   int cps) {
  int s = blockIdx.x;                       // sequence
  int d = threadIdx.x;                      // channel
  float run[NST];
#pragma unroll
  for (int n = 0; n < NST; ++n) run[n] = 0.f;
  for (int c = 0; c < cps; ++c) {
    long o = ((long)(s * cps + c)) * DINC * NST + (long)d * NST;
#pragma unroll
    for (int n = 0; n < NST; ++n) {
      float a = chA[o + n];
      float hh = chH[o + n];
      chH[o + n] = run[n];                  // h0 entering this chunk
      run[n] = a * run[n] + hh;
    }
  }
}

__global__ void scan_final_kernel(const float* __restrict__ uc, const float* __restrict__ dt,
                                  const float* __restrict__ p, const float* __restrict__ A_log,
                                  const float* __restrict__ Dvec, const float* __restrict__ chH,
                                  float* __restrict__ y, int T, int cps, int reverse,
                                  int accumulate) {
  __shared__ float sh[TCH * 2 * NST];       // B|C rows for this chunk: 8 KB
  int gc = blockIdx.x;
  int s = gc / cps, c = gc % cps;
  long seqbase = (long)s * T;
  for (int idx = threadIdx.x; idx < TCH * 2 * NST; idx += blockDim.x) {
    int i = idx >> 5, col = idx & 31;       // col 0-15: B, 16-31: C
    int pos = c * TCH + i;
    int t = reverse ? (T - 1 - pos) : pos;
    sh[idx] = p[(seqbase + t) * PLD + RNK + col];
  }
  __syncthreads();
  int d = threadIdx.x;
  float A[NST], h[NST];
  long o = (long)gc * DINC * NST + (long)d * NST;
#pragma unroll
  for (int n = 0; n < NST; ++n) {
    A[n] = -__expf(A_log[d * NST + n]);
    h[n] = chH[o + n];                      // incoming state
  }
  float Dp = Dvec[d];
  for (int i = 0; i < TCH; ++i) {
    int pos = c * TCH + i;
    int t = reverse ? (T - 1 - pos) : pos;
    long l = seqbase + t;
    float dtv = dt[l * DINC + d];
    float uv = uc[l * DINC + d];
    float du = dtv * uv;
    const float* bs = sh + i * 2 * NST;
    float acc = 0.f;
#pragma unroll
    for (int n = 0; n < NST; ++n) {
      float dA = __expf(dtv * A[n]);
      h[n] = dA * h[n] + du * bs[n];
      acc += h[n] * bs[NST + n];
    }
    acc += uv * Dp;
    float* yp = y + l * DINC + d;
    if (accumulate) *yp += acc; else *yp = acc;
  }
}

// ---------------------------------------------------------------------------
// yg0 = y0*g ; yg1 = y1*g  (elementwise, L*192)
// ---------------------------------------------------------------------------
__global__ void mul2_kernel(const float* __restrict__ y0, const float* __restrict__ y1,
                            const float* __restrict__ g, float* __restrict__ o0,
                            float* __restrict__ o1) {
  long i = (long)blockIdx.x * blockDim.x + threadIdx.x;
  if (i >= (long)LTOT * DINC) return;
  float gv = g[i];
  o0[i] = y0[i] * gv;
  o1[i] = y1[i] * gv;
}

// s = a + b (elementwise, L*96)
__global__ void add_kernel(const float* __restrict__ a, const float* __restrict__ b,
                           float* __restrict__ o) {
  long i = (long)blockIdx.x * blockDim.x + threadIdx.x;
  if (i >= (long)LTOT * CCH) return;
  o[i] = a[i] + b[i];
}

// ---------------------------------------------------------------------------
// Final fusion: w = sigmoid(wpre + fus_b); out(C,L) = osq*w + (1-w)*osl + x
// ---------------------------------------------------------------------------
__global__ void fuse_final_kernel(const float* __restrict__ osq, const float* __restrict__ osl,
                                  const float* __restrict__ wpre, const float* __restrict__ fb,
                                  const float* __restrict__ x, float* __restrict__ out) {
  long idx = (long)blockIdx.x * blockDim.x + threadIdx.x;
  if (idx >= (long)LTOT * CCH) return;
  long l = idx / CCH; int o = (int)(idx % CCH);
  float w = sigm_f(wpre[idx] + fb[o]);
  float v = osq[idx] * w + (1.f - w) * osl[idx] + x[(size_t)o * LTOT + l];
  out[(size_t)o * LTOT + l] = v;
}

// ---------------------------------------------------------------------------
extern "C" void kernel_launch(void* const* d_in, const int* in_sizes, int n_in,
                              void* d_out, int out_size, void* d_ws, size_t ws_size,
                              hipStream_t stream) {
  const float* x        = (const float*)d_in[0];
  const float* ln_g     = (const float*)d_in[1];
  const float* ln_b     = (const float*)d_in[2];
  const float* in_proj  = (const float*)d_in[3];   // (384, 96)
  const float* conv_w   = (const float*)d_in[4];   // (192, 4)
  const float* conv_b   = (const float*)d_in[5];
  const float* xproj_f  = (const float*)d_in[6];   // (38, 192)
  const float* dt_w_f   = (const float*)d_in[7];   // (192, 6)
  const float* dt_b_f   = (const float*)d_in[8];
  const float* A_log_f  = (const float*)d_in[9];   // (192, 16)
  const float* D_f      = (const float*)d_in[10];
  const float* xproj_r  = (const float*)d_in[11];
  const float* dt_w_r   = (const float*)d_in[12];
  const float* dt_b_r   = (const float*)d_in[13];
  const float* A_log_r  = (const float*)d_in[14];
  const float* D_r      = (const float*)d_in[15];
  const float* out_proj = (const float*)d_in[16];  // (96, 192)
  const float* fus_w    = (const float*)d_in[17];  // (96, 96)
  const float* fus_b    = (const float*)d_in[18];
  float* out = (float*)d_out;

  // Workspace layout (floats)
  float* ws = (float*)d_ws;
  size_t off = 0;
  float* xn  = ws + off; off += (size_t)LTOT * CCH;        // (L,96)  later reused as s
  float* xz  = ws + off; off += (size_t)LTOT * 2 * DINC;   // (L,384)
  float* uc  = ws + off; off += (size_t)LTOT * DINC;       // (L,192)
  float* g   = ws + off; off += (size_t)LTOT * DINC;       // (L,192)
  float* pf  = ws + off; off += (size_t)LTOT * PLD;        // (L,48)
  float* pr  = ws + off; off += (size_t)LTOT * PLD;        // (L,48)
  float* dtf = ws + off; off += (size_t)LTOT * DINC;       // (L,192) later reused as yg_sq
  float* dtr = ws + off; off += (size_t)LTOT * DINC;       // (L,192) later reused as yg_sl
  float* ysq = ws + off; off += (size_t)LTOT * DINC;       // (L,192)
  float* ysl = ws + off; off += (size_t)LTOT * DINC;       // (L,192)
  float* osq = ws + off; off += (size_t)LTOT * CCH;        // (L,96)
  float* osl = ws + off; off += (size_t)LTOT * CCH;        // (L,96)
  float* wpre= ws + off; off += (size_t)LTOT * CCH;        // (L,96)
  const int NCHUNK = LTOT / TCH;                           // 128 chunks total either way
  float* chA = ws + off; off += (size_t)NCHUNK * DINC * NST; // chunk A-products
  float* chH = ws + off; off += (size_t)NCHUNK * DINC * NST; // chunk end states -> h0
  (void)ws_size; (void)n_in; (void)in_sizes; (void)out_size;

  auto gemm = [&](const float* X, const float* W, float* Y, int M, int N, int K, int Nvalid) {
    int tiles = (M / 16) * (N / 16);
    int blocks = (tiles + 7) / 8;                  // 8 waves (256 threads) per block
    gemm_wmma_f32<<<blocks, 256, 0, stream>>>(X, W, Y, M, N, K, Nvalid);
  };

  // One bidirectional-scan direction: summary -> prefix -> final
  auto scan_pass = [&](const float* dtb, const float* pb, const float* Al, const float* Dv,
                       float* y, int T, int reverse, int accumulate) {
    int cps = T / TCH;              // chunks per sequence
    int nseq = LTOT / T;
    scan_summary_kernel<<<NCHUNK, DINC, 0, stream>>>(uc, dtb, pb, Al, chA, chH, T, cps, reverse);
    scan_prefix_kernel<<<nseq, DINC, 0, stream>>>(chA, chH, cps);
    scan_final_kernel<<<NCHUNK, DINC, 0, stream>>>(uc, dtb, pb, Al, Dv, chH, y, T, cps,
                                                   reverse, accumulate);
  };

  const long NE = (long)LTOT * DINC;
  const long NC = (long)LTOT * CCH;
  int ebl = (int)((NE + 255) / 256);
  int cbl = (int)((NC + 255) / 256);

  // 1) LayerNorm (C,L) -> (L,C)
  ln_kernel<<<(LTOT + 255) / 256, 256, 0, stream>>>(x, ln_g, ln_b, xn);
  // 2) in_proj: xz = xn @ W^T  (L,96)x(96,384)
  gemm(xn, in_proj, xz, LTOT, 2 * DINC, CCH, 2 * DINC);
  // 3) causal dwconv + silu -> uc ; g = silu(z)
  conv_silu_kernel<<<ebl, 256, 0, stream>>>(xz, conv_w, conv_b, uc, g);
  // 4) xproj GEMMs (Nvalid=38, padded to 48)
  gemm(uc, xproj_f, pf, LTOT, PLD, DINC, RNK + 2 * NST);
  gemm(uc, xproj_r, pr, LTOT, PLD, DINC, RNK + 2 * NST);
  // 5) dt = softplus(...)
  dt_kernel<<<ebl, 256, 0, stream>>>(pf, dt_w_f, dt_b_f, dtf);
  dt_kernel<<<ebl, 256, 0, stream>>>(pr, dt_w_r, dt_b_r, dtr);
  // 6) bidirectional scans (chunk-parallel): full sequence (y_sq), 8 slices (y_sl)
  scan_pass(dtf, pf, A_log_f, D_f, ysq, LTOT, 0, 0);
  scan_pass(dtr, pr, A_log_r, D_r, ysq, LTOT, 1, 1);
  scan_pass(dtf, pf, A_log_f, D_f, ysl, LTOT / 8, 0, 0);
  scan_pass(dtr, pr, A_log_r, D_r, ysl, LTOT / 8, 1, 1);
  // 7) gate: yg = y * g (reuse dtf/dtr buffers; dt no longer needed)
  float* ygsq = dtf;
  float* ygsl = dtr;
  mul2_kernel<<<ebl, 256, 0, stream>>>(ysq, ysl, g, ygsq, ygsl);
  // 8) out_proj GEMMs (L,192)x(192,96)
  gemm(ygsq, out_proj, osq, LTOT, CCH, DINC, CCH);
  gemm(ygsl, out_proj, osl, LTOT, CCH, DINC, CCH);
  // 9) fusion gate: s = osq + osl (reuse xn), wpre = s @ fus_w^T
  float* sbuf = xn;
  add_kernel<<<cbl, 256, 0, stream>>>(osq, osl, sbuf);
  gemm(sbuf, fus_w, wpre, LTOT, CCH, CCH, CCH);
  // 10) final: sigmoid gate blend + residual, write (C,L)
  fuse_final_kernel<<<cbl, 256, 0, stream>>>(osq, osl, wpre, fus_b, x, out);
}